// AMG_encoder_60215441490082
// MI455X (gfx1250) — compile-verified
//
#include <hip/hip_runtime.h>

// ---------------------------------------------------------------------------
// AMG encoder for MI455X (gfx1250, wave32).
//
// Bottleneck: 64 grids x 262144 pts -> random gathers from 128MB of feature
// grids (fully L2-resident, 192MB L2). Bound by L2 gather REQUEST rate, so:
//   * fuse the x-contiguous corner pairs into single 8B loads
//     (8 gather requests per point-grid instead of 16),
//   * keep output stores non-temporal so the 134MB output stream does not
//     evict the L2-resident grids.
// Coordinate transform = V_WMMA_F32_16X16X4_F32 (K=4 homogeneous dim),
// co-executing with the vmem gather stream:
//   A (16x4): lane l holds A[l&15][kb..kb+1], kb=(l>>4)*2.
//     WMMA#1: M rows in A-rows 0..2.  WMMA#2: M rows in A-rows 8..10.
//   B (4x16): VGPR0 = K0 (lanes<16) / K2 (lanes>=16); VGPR1 = K1 / K3.
//     Columns = 16 points' homogeneous (x,y,z,1).
//   D (16x16): VGPR r: lanes 0-15 -> row r, lanes 16-31 -> row 8+r.
//     => every lane ends holding its OWN point's (px,py,pz), no shuffles.
// ---------------------------------------------------------------------------

typedef float v2f  __attribute__((ext_vector_type(2)));
typedef float v8f  __attribute__((ext_vector_type(8)));
typedef float v2fa __attribute__((ext_vector_type(2), aligned(4)));  // 4B-aligned pair load

#define NGRIDS   64
#define NFEAT    2
#define GD       64
#define GH       64
#define GW       64
#define PTS_PER_BLOCK 256

__global__ __launch_bounds__(PTS_PER_BLOCK) void amg_encode_kernel(
    const float* __restrict__ x,      // [N, 3]
    const float* __restrict__ Mmat,   // [G, 4, 4]
    const float* __restrict__ grids,  // [G, F, D, H, W]
    float* __restrict__ out,          // [N, G*F]
    int N)
{
    const int g    = blockIdx.y;
    const int lane = threadIdx.x & 31;
    const int wave = threadIdx.x >> 5;
    const int base = blockIdx.x * PTS_PER_BLOCK + wave * 32;

    const float* __restrict__ Mg = Mmat + g * 16;

    // ---- A fragments (16x4 f32) --------------------------------------------
    const int row = lane & 15;
    const int kb  = (lane >> 4) * 2;
    v2f a1 = {0.0f, 0.0f};                    // M in rows 0..2
    v2f a2 = {0.0f, 0.0f};                    // M in rows 8..10
    if (row < 3) {
        a1.x = Mg[row * 4 + kb];
        a1.y = Mg[row * 4 + kb + 1];
    }
    if (row >= 8 && row < 11) {
        a2.x = Mg[(row - 8) * 4 + kb];
        a2.y = Mg[(row - 8) * 4 + kb + 1];
    }

    // ---- B fragments (4x16 f32): columns = homogeneous points --------------
    const int col = lane & 15;
    int p1c = min(base + col,      N - 1);
    int p2c = min(base + 16 + col, N - 1);
    v2f b1, b2;
    if (lane < 16) {                          // K=0 (x), K=1 (y)
        b1.x = x[p1c * 3 + 0];  b1.y = x[p1c * 3 + 1];
        b2.x = x[p2c * 3 + 0];  b2.y = x[p2c * 3 + 1];
    } else {                                  // K=2 (z), K=3 (w=1)
        b1.x = x[p1c * 3 + 2];  b1.y = 1.0f;
        b2.x = x[p2c * 3 + 2];  b2.y = 1.0f;
    }

    // ---- Transform via V_WMMA_F32_16X16X4_F32 ------------------------------
    v8f c0 = {};
    v8f d1 = __builtin_amdgcn_wmma_f32_16x16x4_f32(false, a1, false, b1,
                                                   (short)0, c0, false, false);
    v8f d2 = __builtin_amdgcn_wmma_f32_16x16x4_f32(false, a2, false, b2,
                                                   (short)0, c0, false, false);

    const float cx = (lane < 16) ? d1[0] : d2[0];
    const float cy = (lane < 16) ? d1[1] : d2[1];
    const float cz = (lane < 16) ? d1[2] : d2[2];
    const int   n  = base + lane;

    // ---- Trilinear setup (align_corners=True, zero padding) ----------------
    const float px = (cx + 1.0f) * 0.5f * (float)(GW - 1);
    const float py = (cy + 1.0f) * 0.5f * (float)(GH - 1);
    const float pz = (cz + 1.0f) * 0.5f * (float)(GD - 1);
    const float fx0 = floorf(px), fy0 = floorf(py), fz0 = floorf(pz);
    const float fx = px - fx0, fy = py - fy0, fz = pz - fz0;
    const int   x0 = (int)fx0, y0 = (int)fy0, z0 = (int)fz0;

    // x-pair fusion: one 8B load at xp covers both x corners.
    const int  xp  = min(max(x0, 0), GW - 2);          // pair base, always in-bounds
    const int  xc0 = min(max(x0,     0), GW - 1);      // clamped corner indices
    const int  xc1 = min(max(x0 + 1, 0), GW - 1);
    const bool sel0 = (xc0 != xp);                     // pick .y lane of the pair?
    const bool sel1 = (xc1 != xp);
    const float wx0 = ((x0 >= 0)  & (x0 < GW))      ? (1.0f - fx) : 0.0f;  // masked x weights
    const float wx1 = ((x0 >= -1) & (x0 <= GW - 2)) ? fx          : 0.0f;

    const float* __restrict__ gf0 = grids + (size_t)g * NFEAT * (GD * GH * GW);
    const float* __restrict__ gf1 = gf0 + (GD * GH * GW);

    float acc0 = 0.0f, acc1 = 0.0f;
#pragma unroll
    for (int dz = 0; dz < 2; ++dz) {
        const int   iz  = z0 + dz;
        const bool  vz  = (iz >= 0) & (iz < GD);
        const int   izc = min(max(iz, 0), GD - 1);
        const float wz  = dz ? fz : (1.0f - fz);
#pragma unroll
        for (int dy = 0; dy < 2; ++dy) {
            const int   iy  = y0 + dy;
            const bool  vy  = (iy >= 0) & (iy < GH);
            const int   iyc = min(max(iy, 0), GH - 1);
            const float wy  = dy ? fy : (1.0f - fy);

            const float wzy = (((vz & vy)) ? 1.0f : 0.0f) * wz * wy;
            const float w0  = wx0 * wzy;
            const float w1  = wx1 * wzy;

            const int rbase = (izc * GH + iyc) * GW + xp;
            const v2fa p0 = *(const v2fa*)(gf0 + rbase);   // feature 0: x-pair (8B gather)
            const v2fa p1 = *(const v2fa*)(gf1 + rbase);   // feature 1: x-pair (8B gather)

            const float v00 = sel0 ? p0.y : p0.x;
            const float v01 = sel1 ? p0.y : p0.x;
            const float v10 = sel0 ? p1.y : p1.x;
            const float v11 = sel1 ? p1.y : p1.x;

            acc0 = fmaf(v01, w1, fmaf(v00, w0, acc0));
            acc1 = fmaf(v11, w1, fmaf(v10, w0, acc1));
        }
    }

    // ---- Store (non-temporal: don't evict the L2-resident grids) -----------
    if (n < N) {
        v2f r = {acc0, acc1};
        v2f* dst = (v2f*)(out + (size_t)n * (NGRIDS * NFEAT) + g * NFEAT);
        __builtin_nontemporal_store(r, dst);
    }
}

extern "C" void kernel_launch(void* const* d_in, const int* in_sizes, int n_in,
                              void* d_out, int out_size, void* d_ws, size_t ws_size,
                              hipStream_t stream) {
    const float* x     = (const float*)d_in[0];   // [N, 3]          f32
    const float* M     = (const float*)d_in[1];   // [64, 4, 4]      f32
    const float* grids = (const float*)d_in[2];   // [64, 2, 64^3]   f32
    float*       out   = (float*)d_out;           // [N, 128]        f32

    const int N = in_sizes[0] / 3;                // 262144
    dim3 grid((N + PTS_PER_BLOCK - 1) / PTS_PER_BLOCK, NGRIDS);
    amg_encode_kernel<<<grid, PTS_PER_BLOCK, 0, stream>>>(x, M, grids, out, N);
}